// GCN_47364899340880
// MI455X (gfx1250) — compile-verified
//
#include <hip/hip_runtime.h>
#include <hip/hip_bf16.h>

#define N_NODES 50000
#define NF 128
#define HDIM 128
#define NCLS 40
#define BN_EPS 1e-5f

typedef __attribute__((ext_vector_type(2))) float v2f;
typedef __attribute__((ext_vector_type(8))) float v8f;

// ---------------------------------------------------------------------------
// Degree / normalization
// ---------------------------------------------------------------------------
__global__ void gcn_deg_init(float* __restrict__ deg, int n) {
    int i = blockIdx.x * blockDim.x + threadIdx.x;
    if (i < n) deg[i] = 1.0f;  // self loop contributes 1
}

__global__ void gcn_deg_edges(const int* __restrict__ dst, float* __restrict__ deg, int nE) {
    int e = blockIdx.x * blockDim.x + threadIdx.x;
    if (e < nE) atomicAdd(&deg[dst[e]], 1.0f);
}

__global__ void gcn_deg_finish(float* __restrict__ dis, int n) {
    int i = blockIdx.x * blockDim.x + threadIdx.x;
    if (i < n) dis[i] = rsqrtf(dis[i]);  // deg >= 1 always (self loops)
}

// ---------------------------------------------------------------------------
// fp32 WMMA GEMM:  C[nrows x 128] = A[nrows x 128] @ B[128 x 128]
// Block = 256 threads = 8 waves. Block covers 16 rows x 128 cols; each wave
// owns one 16x16 C tile and iterates K in steps of 4 with
// V_WMMA_F32_16X16X4_F32.
//
// Operand layouts (ISA 7.12.2, wave32):
//   A 16x4 f32 : lane m=lane&15 holds row m; vgpr j holds K = 2*(lane>>4)+j
//   B 4x16 f32 : lane n=lane&15 holds col n; vgpr j holds K = 2*(lane>>4)+j
//   C 16x16 f32: vgpr r, lanes 0-15 -> M=r, lanes 16-31 -> M=8+r; N=lane&15
// ---------------------------------------------------------------------------
__global__ __launch_bounds__(256) void gcn_gemm128_wmma(
    const float* __restrict__ A, const float* __restrict__ B,
    float* __restrict__ C, int nrows) {
    const int lane = threadIdx.x & 31;
    const int wave = threadIdx.x >> 5;      // 0..7
    const int rb   = blockIdx.x * 16;       // row base (nrows % 16 == 0)
    const int ct   = wave * 16;             // column tile base
    const int m    = lane & 15;
    const int kh   = lane >> 4;             // 0 or 1

    v8f acc = {};
    const float* arow = A + (size_t)(rb + m) * 128;
#pragma unroll
    for (int kb = 0; kb < 128; kb += 4) {
        const int k0 = kb + 2 * kh;
        v2f av; av.x = arow[k0];
                av.y = arow[k0 + 1];
        v2f bv; bv.x = B[(size_t)k0 * 128 + ct + m];
                bv.y = B[(size_t)(k0 + 1) * 128 + ct + m];
        acc = __builtin_amdgcn_wmma_f32_16x16x4_f32(
            /*neg_a=*/false, av, /*neg_b=*/false, bv,
            /*c_mod=*/(short)0, acc, /*reuse_a=*/false, /*reuse_b=*/false);
    }
    const int rowoff = kh * 8;
#pragma unroll
    for (int r = 0; r < 8; ++r)
        C[(size_t)(rb + rowoff + r) * 128 + ct + m] = acc[r];
}

// ---------------------------------------------------------------------------
// agg[i][c] = hw[i][c] * dis[i]^2 + bias[c]   (self-loop term + conv bias)
// ---------------------------------------------------------------------------
__global__ void gcn_agg_init(const float* __restrict__ hw, const float* __restrict__ dis,
                             const float* __restrict__ bias, float* __restrict__ agg,
                             int total) {
    int idx = blockIdx.x * blockDim.x + threadIdx.x;
    if (idx >= total) return;
    int node = idx >> 7;
    int c    = idx & 127;
    float d  = dis[node];
    agg[idx] = hw[idx] * d * d + bias[c];
}

// ---------------------------------------------------------------------------
// Edge scatter: one wave per edge, 4 features per lane (float4 gather,
// 4x global_atomic_add_f32 scatter). agg is L2-resident (25.6MB << 192MB).
// ---------------------------------------------------------------------------
__global__ __launch_bounds__(256) void gcn_agg_edges(
    const int* __restrict__ src, const int* __restrict__ dst,
    const float* __restrict__ hw, const float* __restrict__ dis,
    float* __restrict__ agg, int nE) {
    int e = blockIdx.x * 8 + (threadIdx.x >> 5);
    if (e >= nE) return;
    int lane = threadIdx.x & 31;
    int s = src[e], d = dst[e];
    float nrm = dis[s] * dis[d];
    const float4 v = *(const float4*)(hw + (size_t)s * 128 + lane * 4);
    float* ap = agg + (size_t)d * 128 + lane * 4;
    atomicAdd(ap + 0, v.x * nrm);
    atomicAdd(ap + 1, v.y * nrm);
    atomicAdd(ap + 2, v.z * nrm);
    atomicAdd(ap + 3, v.w * nrm);
}

// ---------------------------------------------------------------------------
// BatchNorm statistics (sum, sumsq per channel). Channel = idx & 127 is
// invariant per thread because stride (gridDim*256) is a multiple of 128.
// LDS accumulation -> one global atomic per channel per block.
// ---------------------------------------------------------------------------
__global__ void gcn_stats_zero(float* __restrict__ stats) {
    if (threadIdx.x < 256) stats[threadIdx.x] = 0.0f;
}

__global__ __launch_bounds__(256) void gcn_bn_stats(
    const float* __restrict__ X, float* __restrict__ stats, int total) {
    __shared__ float ssum[128];
    __shared__ float ssq[128];
    if (threadIdx.x < 128) { ssum[threadIdx.x] = 0.0f; ssq[threadIdx.x] = 0.0f; }
    __syncthreads();
    const int c = threadIdx.x & 127;
    float lsum = 0.0f, lsq = 0.0f;
    const int stride = gridDim.x * blockDim.x;
    for (int idx = blockIdx.x * blockDim.x + threadIdx.x; idx < total; idx += stride) {
        float v = X[idx];
        lsum += v;
        lsq  += v * v;
    }
    atomicAdd(&ssum[c], lsum);
    atomicAdd(&ssq[c], lsq);
    __syncthreads();
    if (threadIdx.x < 128) {
        atomicAdd(&stats[threadIdx.x],       ssum[threadIdx.x]);
        atomicAdd(&stats[128 + threadIdx.x], ssq[threadIdx.x]);
    }
}

// ---------------------------------------------------------------------------
// y = relu(g * (x - mean) * rsqrt(var + eps) + beta)
// ---------------------------------------------------------------------------
__global__ void gcn_bn_relu(const float* __restrict__ X, const float* __restrict__ stats,
                            const float* __restrict__ g, const float* __restrict__ be,
                            float* __restrict__ Y, int total) {
    int idx = blockIdx.x * blockDim.x + threadIdx.x;
    if (idx >= total) return;
    int c = idx & 127;
    const float invn = 1.0f / (float)N_NODES;
    float mean = stats[c] * invn;
    float var  = stats[128 + c] * invn - mean * mean;
    float y = g[c] * (X[idx] - mean) * rsqrtf(var + BN_EPS) + be[c];
    Y[idx] = fmaxf(y, 0.0f);
}

// ---------------------------------------------------------------------------
// logits[r][c] = relu(dot(h[r], fcW[:,c]) + fcb[c]) ;  fcW is (128, 40)
// ---------------------------------------------------------------------------
__global__ void gcn_fc_relu(const float* __restrict__ h, const float* __restrict__ Wf,
                            const float* __restrict__ bf, float* __restrict__ logits,
                            int nrows) {
    int idx = blockIdx.x * blockDim.x + threadIdx.x;
    if (idx >= nrows * NCLS) return;
    int r = idx / NCLS;
    int c = idx - r * NCLS;
    const float* hr = h + (size_t)r * 128;
    float s = bf[c];
#pragma unroll 8
    for (int k = 0; k < 128; ++k) s = fmaf(hr[k], Wf[k * NCLS + c], s);
    logits[idx] = fmaxf(s, 0.0f);
}

// ---------------------------------------------------------------------------
// log_softmax over 40 classes: one wave32 per row, lane handles classes
// {lane, lane+32}; max/sum via shuffle-xor reductions.
// ---------------------------------------------------------------------------
__global__ __launch_bounds__(256) void gcn_log_softmax(
    const float* __restrict__ logits, float* __restrict__ out, int nrows) {
    int row = blockIdx.x * 8 + (threadIdx.x >> 5);
    if (row >= nrows) return;
    int lane = threadIdx.x & 31;
    const float* lr = logits + (size_t)row * NCLS;
    float v0 = lr[lane];  // lane in [0,32) < 40, always valid
    bool has2 = (lane + 32) < NCLS;
    float v1 = has2 ? lr[lane + 32] : -3.0e38f;
    float m = fmaxf(v0, v1);
#pragma unroll
    for (int off = 16; off > 0; off >>= 1) m = fmaxf(m, __shfl_xor(m, off, 32));
    float s = __expf(v0 - m) + (has2 ? __expf(v1 - m) : 0.0f);
#pragma unroll
    for (int off = 16; off > 0; off >>= 1) s += __shfl_xor(s, off, 32);
    float ls = __logf(s);
    out[(size_t)row * NCLS + lane] = v0 - m - ls;
    if (has2) out[(size_t)row * NCLS + lane + 32] = v1 - m - ls;
}

// ---------------------------------------------------------------------------
// Orchestration
// ---------------------------------------------------------------------------
extern "C" void kernel_launch(void* const* d_in, const int* in_sizes, int n_in,
                              void* d_out, int out_size, void* d_ws, size_t ws_size,
                              hipStream_t stream) {
    const int n = N_NODES;
    const int E = in_sizes[1] / 2;

    const float* x    = (const float*)d_in[0];
    const int*   src  = (const int*)d_in[1];
    const int*   dst  = src + E;
    const float* Ws[3]  = {(const float*)d_in[2], (const float*)d_in[4], (const float*)d_in[6]};
    const float* bs[3]  = {(const float*)d_in[3], (const float*)d_in[5], (const float*)d_in[7]};
    const float* gs[3]  = {(const float*)d_in[8], (const float*)d_in[10], (const float*)d_in[12]};
    const float* bes[3] = {(const float*)d_in[9], (const float*)d_in[11], (const float*)d_in[13]};
    const float* fcW = (const float*)d_in[14];
    const float* fcb = (const float*)d_in[15];
    float* out = (float*)d_out;

    // workspace layout (floats)
    float* ws    = (float*)d_ws;
    float* dis   = ws;                          // n
    float* hw    = dis + n;                     // n*128
    float* agg   = hw + (size_t)n * HDIM;       // n*128
    float* act   = agg + (size_t)n * HDIM;      // n*128
    float* stats = act + (size_t)n * HDIM;      // 256

    const int total = n * HDIM;

    // degree + symmetric normalization
    gcn_deg_init<<<(n + 255) / 256, 256, 0, stream>>>(dis, n);
    gcn_deg_edges<<<(E + 255) / 256, 256, 0, stream>>>(dst, dis, E);
    gcn_deg_finish<<<(n + 255) / 256, 256, 0, stream>>>(dis, n);

    const float* hin = x;
    for (int l = 0; l < 3; ++l) {
        gcn_gemm128_wmma<<<n / 16, 256, 0, stream>>>(hin, Ws[l], hw, n);
        gcn_agg_init<<<(total + 255) / 256, 256, 0, stream>>>(hw, dis, bs[l], agg, total);
        gcn_agg_edges<<<(E + 7) / 8, 256, 0, stream>>>(src, dst, hw, dis, agg, E);
        gcn_stats_zero<<<1, 256, 0, stream>>>(stats);
        gcn_bn_stats<<<1024, 256, 0, stream>>>(agg, stats, total);
        gcn_bn_relu<<<(total + 255) / 256, 256, 0, stream>>>(agg, stats, gs[l], bes[l], act, total);
        hin = act;
    }

    float* logits = out + (size_t)n * NCLS;  // second output of the tuple
    gcn_fc_relu<<<(n * NCLS + 255) / 256, 256, 0, stream>>>(act, fcW, fcb, logits, n);
    gcn_log_softmax<<<(n + 7) / 8, 256, 0, stream>>>(logits, out, n);
}